// Loss_12326556139840
// MI455X (gfx1250) — compile-verified
//
#include <hip/hip_runtime.h>
#include <hip/hip_bf16.h>
#include <math.h>

typedef __attribute__((ext_vector_type(2))) float v2f;
typedef __attribute__((ext_vector_type(8))) float v8f;

#define EPSF      1e-7f
#define SCALEF    (1.0f / 7.0f)
#define CPB       256     // cells per block
#define FPC       30      // floats per cell

// Elementwise IoU of cxcywh boxes, faithful to the reference's eps placement.
__device__ __forceinline__ float pair_iou(const float* b1, const float* b2) {
    float b1x1 = b1[0] - b1[2] * 0.5f;
    float b1x2 = b1[0] + b1[2] * 0.5f;
    float b1y1 = b1[1] - b1[3] * 0.5f;
    float b1y2 = b1[1] + b1[3] * 0.5f;
    float b2x1 = b2[0] - b2[2] * 0.5f;
    float b2x2 = b2[0] + b2[2] * 0.5f;
    float b2y1 = b2[1] - b2[3] * 0.5f;
    float b2y2 = b2[1] + b2[3] * 0.5f;
    float iw = fmaxf(fminf(b1x2, b2x2) - fmaxf(b1x1, b2x1), 0.0f);
    float ih = fmaxf(fminf(b1y2, b2y2) - fmaxf(b1y1, b2y1), 0.0f);
    float inter = iw * ih + EPSF;
    float w1 = b1x2 - b1x1;
    float h1 = b1y2 - b1y1 + EPSF;
    float w2 = b2x2 - b2x1;
    float h2 = b2y2 - b2y1 + EPSF;
    float uni = w1 * h1 + w2 * h2 - inter + EPSF;
    return inter / uni;
}

// Kernel 1: one thread per 7x7 cell; block stages 256 cells (30 f32 each) of
// both tensors through LDS with b128 loads, computes the per-cell loss,
// reduces the block to one partial (deterministic tree), writes to ws.
__global__ __launch_bounds__(256) void yolo_loss_partial(
    const float* __restrict__ pre, const float* __restrict__ tgt,
    float* __restrict__ partial) {
    __shared__ __align__(16) float sp[CPB * FPC];
    __shared__ __align__(16) float st[CPB * FPC];
    __shared__ float wsum[8];

    const size_t base = (size_t)blockIdx.x * (CPB * FPC);
    const float4* gp = (const float4*)(pre + base);
    const float4* gt = (const float4*)(tgt + base);
    float4* lp = (float4*)sp;
    float4* lt = (float4*)st;
    // 256*30/4 = 1920 float4 per tensor, cooperatively staged (coalesced b128)
    #pragma unroll
    for (int j = threadIdx.x; j < (CPB * FPC / 4); j += 256) {
        lp[j] = gp[j];
        lt[j] = gt[j];
    }
    __syncthreads();

    const float* p = sp + threadIdx.x * FPC;
    const float* t = st + threadIdx.x * FPC;

    float iou0 = pair_iou(p, t);
    float iou1 = pair_iou(p + 5, t + 5);
    // jnp.argmax / argmin return the FIRST index on ties:
    int resp    = (iou1 > iou0) ? 1 : 0;
    int nonresp = (iou0 <= iou1) ? 0 : 1;

    bool obj   = t[4] > 0.0f;
    bool noobj = t[4] == 0.0f;

    // fractional xy of the non-responsible target box's cell position
    float xnr = t[5 * nonresp + 0];
    float ynr = t[5 * nonresp + 1];
    float fx = (xnr - (ceilf(xnr / SCALEF) - 1.0f) * SCALEF) / SCALEF;
    float fy = (ynr - (ceilf(ynr / SCALEF) - 1.0f) * SCALEF) / SCALEF;

    float lxy = 0.f, lwh = 0.f, lconf = 0.f, lnoconf = 0.f, lclass = 0.f;
    #pragma unroll
    for (int i = 0; i < 2; ++i) {
        float px = p[5 * i + 0], py = p[5 * i + 1];
        float pw = p[5 * i + 2], ph = p[5 * i + 3], pc = p[5 * i + 4];
        float tx = t[5 * i + 0], ty = t[5 * i + 1];
        float tw = t[5 * i + 2], th = t[5 * i + 3], tc = t[5 * i + 4];
        float ncf = (i == nonresp) ? 0.0f : tc;
        float nx  = (i == resp) ? fx : tx;
        float ny  = (i == resp) ? fy : ty;
        bool mr = (i == resp) && obj;
        bool mn = (i != resp) && obj;
        if (mr) {
            float dx = px - nx, dy = py - ny;
            lxy += dx * dx + dy * dy;
            float dw = sqrtf(pw + EPSF) - sqrtf(tw + EPSF);
            float dh = sqrtf(ph + EPSF) - sqrtf(th + EPSF);
            lwh += dw * dw + dh * dh;
            float dc = pc - ncf;
            lconf += dc * dc;
        }
        if (mn) { float dc = pc - ncf; lnoconf += dc * dc; }
        if (noobj) { float dc = pc - tc; lnoconf += dc * dc; }
    }
    if (obj) {
        #pragma unroll
        for (int j = 10; j < 30; ++j) {
            float d = p[j] - t[j];
            lclass += d * d;
        }
    }
    float v = 5.0f * (lxy + lwh) + lclass + lconf + 0.5f * lnoconf;

    // wave32 reduction, then 8 wave sums folded by thread 0 (deterministic)
    #pragma unroll
    for (int m = 16; m; m >>= 1) v += __shfl_xor(v, m, 32);
    int lane = threadIdx.x & 31;
    int wid  = threadIdx.x >> 5;
    if (lane == 0) wsum[wid] = v;
    __syncthreads();
    if (threadIdx.x == 0) {
        float s = 0.f;
        #pragma unroll
        for (int w = 0; w < 8; ++w) s += wsum[w];
        partial[blockIdx.x] = s;
    }
}

// Kernel 2: one wave32. Each lane strided-sums the block partials (n % 32 == 0
// keeps EXEC uniform), then a single V_WMMA_F32_16X16X4_F32 with A = ones
// folds all 32 lane partials: D[m,n] = sum_k B[k,n], so sum_{n<16} D[0,n] is
// the total. Four xor-shuffles finish; lane 0 writes loss / B.
__global__ __launch_bounds__(32) void yolo_loss_finalize(
    const float* __restrict__ ws, float* __restrict__ out, int n, float invB) {
    int lane = threadIdx.x;
    float s = 0.f;
    for (int i = lane; i < n; i += 32) s += ws[i];

    v2f a = {1.0f, 1.0f};   // A 16x4 all-ones
    v2f b = {s, 0.0f};      // each lane's partial at a unique (K,N); rest zero
    v8f c = {};
    c = __builtin_amdgcn_wmma_f32_16x16x4_f32(
        /*neg_a=*/false, a, /*neg_b=*/false, b,
        /*c_mod=*/(short)0, c, /*reuse_a=*/false, /*reuse_b=*/false);

    float r = c[0];         // (M=0, N=lane) for lanes 0..15
    r += __shfl_xor(r, 1, 32);
    r += __shfl_xor(r, 2, 32);
    r += __shfl_xor(r, 4, 32);
    r += __shfl_xor(r, 8, 32);
    if (lane == 0) out[0] = r * invB;
}

extern "C" void kernel_launch(void* const* d_in, const int* in_sizes, int n_in,
                              void* d_out, int out_size, void* d_ws, size_t ws_size,
                              hipStream_t stream) {
    (void)n_in; (void)out_size; (void)ws_size;
    const float* pre = (const float*)d_in[0];
    const float* tgt = (const float*)d_in[1];
    float* out = (float*)d_out;
    float* ws  = (float*)d_ws;

    int B = in_sizes[0] / (7 * 7 * 30);     // 16384
    int cells = B * 49;                     // 802816
    int blocks = cells / CPB;               // 3136 (exact; 3136 % 32 == 0)

    yolo_loss_partial<<<blocks, CPB, 0, stream>>>(pre, tgt, ws);
    yolo_loss_finalize<<<1, 32, 0, stream>>>(ws, out, blocks, 1.0f / (float)B);
}